// GatingNetwork_15006615734190
// MI455X (gfx1250) — compile-verified
//
#include <hip/hip_runtime.h>
#include <math.h>

#define D_MODEL     2048
#define NUM_EXPERTS 64
#define TOP_K       2
#define N_TOKENS    16384
#define TOK_PER_WG  32
#define KC          64           // K-chunk streamed through LDS (double-buffered)
#define LDX         (KC + 4)     // +4 dwords pad (done by TDM pad_enable) -> conflict-free ds_load_b64
#define NCHUNK      (D_MODEL / KC)
#define XCH         (TOK_PER_WG * LDX)     // floats per x chunk buffer
#define WCH         (NUM_EXPERTS * LDX)    // floats per W chunk buffer

typedef __attribute__((ext_vector_type(2))) float v2f;
typedef __attribute__((ext_vector_type(8))) float v8f;
typedef __attribute__((ext_vector_type(4))) unsigned int v4u;
typedef __attribute__((ext_vector_type(4))) int v4i;
typedef __attribute__((ext_vector_type(8))) int v8i;

// Issue one TDM 2-D tile load: rows x KC f32 elements, row stride D_MODEL,
// into LDS at lds_off_bytes with a 4-dword pad after every 64 dwords
// (hardware padding -> LDS row stride of LDX dwords).
__device__ __forceinline__ void tdm_load_2d(const float* gsrc, int rows,
                                            unsigned lds_off_bytes) {
    unsigned long long ga = (unsigned long long)(uintptr_t)gsrc;
    // D# group 0: count=1 | lds_addr | global_addr[56:0] | type=2
    v4u g0 = { 1u,
               lds_off_bytes,
               (unsigned)ga,
               (unsigned)((ga >> 32) & 0x1FFFFFFull) | (2u << 30) };
    // D# group 1
    v8i g1;
    g1[0] = (2 << 16)        // data_size = 4 bytes
          | (1 << 20)        // pad_enable
          | (5 << 22)        // pad_interval: 64 dwords before each pad
          | (3 << 25);       // pad_amount: 4 dwords
    g1[1] = (int)(((unsigned)D_MODEL & 0xFFFFu) << 16);          // tensor_dim0[15:0]
    g1[2] = (int)(((unsigned)D_MODEL >> 16) & 0xFFFFu)           // tensor_dim0[31:16]
          | ((rows & 0xFFFF) << 16);                             // tensor_dim1[15:0]
    g1[3] = ((KC & 0xFFFF) << 16);                               // tile_dim0 = KC
    g1[4] = (rows & 0xFFFF);                                     // tile_dim1 = rows (tile_dim2=0)
    g1[5] = (int)D_MODEL;                                        // tensor_dim0_stride[31:0]
    g1[6] = 0;                                                   // stride hi / dim1_stride lo
    g1[7] = 0;
    v4i z4 = { 0, 0, 0, 0 };            // groups 2/3 unused (2-D tile)
    v8i z8 = { 0, 0, 0, 0, 0, 0, 0, 0 };
    __builtin_amdgcn_tensor_load_to_lds(g0, g1, z4, z4, z8, 0);
}

__global__ __launch_bounds__(256)
void gating_wmma_kernel(const float* __restrict__ x,
                        const float* __restrict__ W,
                        const float* __restrict__ b,
                        float* __restrict__ out_w,   // [N,2] top-k weights
                        float* __restrict__ out_i,   // [N,2] top-k indices (as float)
                        float* __restrict__ out_m,   // [N,64] one-hot union mask
                        float* __restrict__ out_p)   // [N,64] probabilities
{
    __shared__ float lds_x[2 * XCH];                       // 17408 B
    __shared__ float lds_w[2 * WCH];                       // 34816 B
    __shared__ float lds_logits[TOK_PER_WG * NUM_EXPERTS]; //  8192 B

    const int tid  = threadIdx.x;
    const int wave = tid >> 5;
    const int lane = tid & 31;
    const int h    = lane >> 4;   // lane half (K-group select per ISA A/B layout)
    const int mn   = lane & 15;   // row (A: token) / col (B: expert)
    const int tt   = wave >> 2;   // token tile 0..1
    const int et   = wave & 3;    // expert tile 0..3
    const int tok0 = blockIdx.x * TOK_PER_WG;

    const unsigned xbase = (unsigned)(uintptr_t)&lds_x[0];
    const unsigned wbase = (unsigned)(uintptr_t)&lds_w[0];

    v8f acc0 = {};
    v8f acc1 = {};

    const float* xg = x + (size_t)tok0 * D_MODEL;

    // Prologue: prefetch chunk 0 into buffer 0 (TDM, wave 0 only; EXEC ignored by TDM)
    if (wave == 0) {
        tdm_load_2d(xg, TOK_PER_WG, xbase);
        tdm_load_2d(W,  NUM_EXPERTS, wbase);
    }

    for (int c = 0; c < NCHUNK; ++c) {
        const int cur = c & 1;
        if (wave == 0) {
            if (c + 1 < NCHUNK) {
                // prefetch next chunk into the other buffer, then wait until
                // only those 2 TDM ops are outstanding (in-order completion
                // => current chunk's tiles have landed in LDS)
                tdm_load_2d(xg + (size_t)(c + 1) * KC, TOK_PER_WG,
                            xbase + (unsigned)((cur ^ 1) * XCH * 4));
                tdm_load_2d(W + (size_t)(c + 1) * KC, NUM_EXPERTS,
                            wbase + (unsigned)((cur ^ 1) * WCH * 4));
                __builtin_amdgcn_s_wait_tensorcnt((short)2);
            } else {
                __builtin_amdgcn_s_wait_tensorcnt((short)0);
            }
        }
        __syncthreads();   // publish chunk `cur` to all waves

        // ---- WMMA inner loop: D(16x16) += A(16x4 tokens) x B(4x16 experts) ----
        // A layout per ISA: VGPR v, lane half h -> K = kk + 2*h + v (pair -> ds_load_b64)
        const float* ax = &lds_x[cur * XCH + (tt * 16 + mn) * LDX + 2 * h];
        const float* bx = &lds_w[cur * WCH + (et * 16 + mn) * LDX + 2 * h];
        #pragma unroll
        for (int kk = 0; kk < KC; kk += 8) {
            v2f a0 = *(const v2f*)(ax + kk);
            v2f b0 = *(const v2f*)(bx + kk);
            acc0 = __builtin_amdgcn_wmma_f32_16x16x4_f32(false, a0, false, b0,
                                                         (short)0, acc0, false, false);
            v2f a1 = *(const v2f*)(ax + kk + 4);
            v2f b1 = *(const v2f*)(bx + kk + 4);
            acc1 = __builtin_amdgcn_wmma_f32_16x16x4_f32(false, a1, false, b1,
                                                         (short)0, acc1, false, false);
        }
        __syncthreads();   // all waves done reading `cur` before TDM reuses it
    }

    // ---- spill logits tile to LDS: C/D layout: VGPR r, half h -> M = r + 8h ----
    #pragma unroll
    for (int r = 0; r < 8; ++r) {
        lds_logits[(tt * 16 + r + 8 * h) * NUM_EXPERTS + et * 16 + mn] = acc0[r] + acc1[r];
    }
    __syncthreads();

    // ---- phase 2: softmax + top-2 per token, one wave per 4 tokens ----
    const float bias0 = b[lane];
    const float bias1 = b[lane + 32];
    #pragma unroll
    for (int j = 0; j < 4; ++j) {
        const int lt  = wave * 4 + j;
        const int tok = tok0 + lt;

        float v0 = lds_logits[lt * NUM_EXPERTS + lane] + bias0;
        float v1 = lds_logits[lt * NUM_EXPERTS + lane + 32] + bias1;

        float m = fmaxf(v0, v1);
        #pragma unroll
        for (int off = 16; off > 0; off >>= 1)
            m = fmaxf(m, __shfl_xor(m, off, 32));

        float e0 = __expf(v0 - m), e1 = __expf(v1 - m);
        float s = e0 + e1;
        #pragma unroll
        for (int off = 16; off > 0; off >>= 1)
            s += __shfl_xor(s, off, 32);
        float inv = 1.0f / s;
        float p0 = e0 * inv, p1 = e1 * inv;

        out_p[(size_t)tok * NUM_EXPERTS + lane]      = p0;
        out_p[(size_t)tok * NUM_EXPERTS + lane + 32] = p1;

        // top-1 (tie-break: lower index, matching lax.top_k)
        float cv; int ci;
        if (p0 >= p1) { cv = p0; ci = lane; } else { cv = p1; ci = lane + 32; }
        #pragma unroll
        for (int off = 16; off > 0; off >>= 1) {
            float ov = __shfl_xor(cv, off, 32);
            int   oi = __shfl_xor(ci, off, 32);
            if (ov > cv || (ov == cv && oi < ci)) { cv = ov; ci = oi; }
        }
        const int   i1  = ci;
        const float p1v = cv;

        // top-2: exclude i1, reduce again
        float c0 = (lane == i1)      ? -1.0f : p0;
        float c1 = (lane + 32 == i1) ? -1.0f : p1;
        if (c0 >= c1) { cv = c0; ci = lane; } else { cv = c1; ci = lane + 32; }
        #pragma unroll
        for (int off = 16; off > 0; off >>= 1) {
            float ov = __shfl_xor(cv, off, 32);
            int   oi = __shfl_xor(ci, off, 32);
            if (ov > cv || (ov == cv && oi < ci)) { cv = ov; ci = oi; }
        }
        const int   i2  = ci;
        const float p2v = cv;

        const float wn = 1.0f / (p1v + p2v);
        if (lane == 0) {
            out_w[(size_t)tok * TOP_K]     = p1v * wn;
            out_w[(size_t)tok * TOP_K + 1] = p2v * wn;
            out_i[(size_t)tok * TOP_K]     = (float)i1;
            out_i[(size_t)tok * TOP_K + 1] = (float)i2;
        }
        out_m[(size_t)tok * NUM_EXPERTS + lane]      = (lane == i1 || lane == i2) ? 1.0f : 0.0f;
        out_m[(size_t)tok * NUM_EXPERTS + lane + 32] = (lane + 32 == i1 || lane + 32 == i2) ? 1.0f : 0.0f;
    }
}

extern "C" void kernel_launch(void* const* d_in, const int* in_sizes, int n_in,
                              void* d_out, int out_size, void* d_ws, size_t ws_size,
                              hipStream_t stream) {
    const float* x = (const float*)d_in[0];
    const float* W = (const float*)d_in[1];
    const float* b = (const float*)d_in[2];

    float* out   = (float*)d_out;
    float* out_w = out;                                          // [N,2]
    float* out_i = out + (size_t)N_TOKENS * TOP_K;               // [N,2]
    float* out_m = out + (size_t)2 * N_TOKENS * TOP_K;           // [N,64]
    float* out_p = out_m + (size_t)N_TOKENS * NUM_EXPERTS;       // [N,64]

    dim3 grid(N_TOKENS / TOK_PER_WG);   // 512 workgroups
    dim3 block(256);                    // 8 wave32
    gating_wmma_kernel<<<grid, block, 0, stream>>>(x, W, b, out_w, out_i, out_m, out_p);
}